// PatchedSegmentationMapPredictor_10376640987189
// MI455X (gfx1250) — compile-verified
//
#include <hip/hip_runtime.h>

#define PATCH 7

typedef __attribute__((ext_vector_type(2))) float v2f;
typedef __attribute__((ext_vector_type(8))) float v8f;

// ---------------------------------------------------------------------------
// One f32 GEMM layer of the MLP: out[M,N] = act(A[M,K] @ W[K,N] + bias)
// One wave32 computes a 16x64 C strip (4 adjacent 16x16 tiles) with
// V_WMMA_F32_16X16X4_F32, reusing one A fragment across 4 WMMAs per K-step.
//
// VGPR layouts (CDNA5 ISA 7.12.2, wave32):
//   A 16x4:  lane<16 -> m=lane,     vgpr{0,1} = K{0,1}
//            lane>=16 -> m=lane-16, vgpr{0,1} = K{2,3}
//   B 4x16:  vgpr r -> row (r + 2*half), col = lane&15
//   C 16x16: vgpr r -> row (r + 8*half), col = lane&15
//
// KF/NF > 0: compile-time dims -> all loads use immediate offsets off two
// base pointers (A span K*4B, B span K*N*4B both fit signed-24-bit IOFFSET).
// KF==NF==0: runtime-dim fallback. Requires Ncol % 64 == 0, K % 4 == 0.
// ---------------------------------------------------------------------------
template<int KF, int NF>
__global__ void __launch_bounds__(256)
mlp_layer_wmma(const float* __restrict__ A, const float* __restrict__ Wt,
               const float* __restrict__ bias, float* __restrict__ out,
               int M, int Krt, int Nrt, int relu)
{
    const int K    = (KF > 0) ? KF : Krt;
    const int Ncol = (NF > 0) ? NF : Nrt;
    constexpr int NT = 4;                       // 16x16 tiles per wave (N dir)

    const int lane   = threadIdx.x & 31;
    const int wave   = (int)((blockIdx.x * blockDim.x + threadIdx.x) >> 5);
    const int tilesN = Ncol / (16 * NT);
    const int tm     = wave / tilesN;
    const int tn     = wave - tm * tilesN;
    const int row0   = tm << 4;
    const int col0   = tn * (16 * NT);
    if (row0 >= M) return;

    const int half = lane >> 4;                 // 0: lanes 0-15, 1: 16-31
    const int l15  = lane & 15;

    // clamp A row for ragged M (stores guarded below)
    const int ar = (row0 + l15) < M ? (row0 + l15) : (M - 1);
    const float* Arow  = A  + (size_t)ar * K + 2 * half;              // + kb
    const float* Bbase = Wt + (size_t)(2 * half) * Ncol + col0 + l15; // + kb*N

    v8f acc[NT] = {};
#pragma unroll 4
    for (int kb = 0; kb < K; kb += 4) {
        v2f a;
        a.x = Arow[kb + 0];                     // K = kb + 2*half
        a.y = Arow[kb + 1];                     // K = kb + 2*half + 1
#pragma unroll
        for (int t = 0; t < NT; ++t) {
            v2f b;
            b.x = Bbase[(size_t)(kb + 0) * Ncol + t * 16];  // row kb+2*half
            b.y = Bbase[(size_t)(kb + 1) * Ncol + t * 16];  // row kb+2*half+1
            acc[t] = __builtin_amdgcn_wmma_f32_16x16x4_f32(
                /*neg_a=*/false, a, /*neg_b=*/false, b,
                /*c_mod=*/(short)0, acc[t], /*reuse_a=*/false, /*reuse_b=*/false);
        }
    }

#pragma unroll
    for (int t = 0; t < NT; ++t) {
        const int ncol = col0 + t * 16 + l15;
        const float bb = bias[ncol];
#pragma unroll
        for (int r = 0; r < 8; ++r) {
            const int m = row0 + r + 8 * half;
            float v = acc[t][r] + bb;
            if (relu) v = fmaxf(v, 0.0f);
            if (m < M) out[(size_t)m * Ncol + ncol] = v;
        }
    }
}

// ---------------------------------------------------------------------------
// Patch scoring: one wave32 per query.
//   logits[n,p] = dot(feature_map[b, py, px, :], qe[n, :])
//   indices[n,p] = (b, py, px, q_in_image)
// qe preloaded as float4 chunks per lane; wave reduce via shfl_xor (ds path).
// DF > 0: compile-time D -> preload/dot loops fully unroll.
// ---------------------------------------------------------------------------
template<int DF>
__global__ void __launch_bounds__(256)
patch_kernel(const float* __restrict__ fm, const float* __restrict__ qe,
             const float* __restrict__ qpos, const int* __restrict__ shapes,
             float* __restrict__ logits, int* __restrict__ indices,
             int B, int Q, int Drt)
{
    const int D    = (DF > 0) ? DF : Drt;
    const int lane = threadIdx.x & 31;
    const int n    = (int)((blockIdx.x * blockDim.x + threadIdx.x) >> 5);
    const int Ntot = B * Q;
    if (n >= Ntot) return;

    const int b  = n / Q;
    const int qi = n - b * Q;
    const int H  = shapes[2 * b + 0];
    const int W  = shapes[2 * b + 1];

    const float fx = qpos[2 * n + 0];   // normalized x
    const float fy = qpos[2 * n + 1];   // normalized y
    const int cy = (int)(fy * (float)H);
    const int cx = (int)(fx * (float)W);

    // preload query embedding: float4 per lane, 128-float stripes
    const float* qv = qe + (size_t)n * D;
    float4 qreg[8];
    int nch = 0;
#pragma unroll
    for (int d = lane * 4; d < D; d += 128) qreg[nch++] = *(const float4*)(qv + d);

    const size_t fm_b = (size_t)b * (size_t)H * (size_t)W * (size_t)D;

    // COO indices: lanes stride the 49 cells, 16B int4 stores
    for (int p = lane; p < PATCH * PATCH; p += 32) {
        int dy = p / PATCH - PATCH / 2;
        int dx = p % PATCH - PATCH / 2;
        int py = cy + dy; py = py < 0 ? 0 : (py > H - 1 ? H - 1 : py);
        int px = cx + dx; px = px < 0 ? 0 : (px > W - 1 ? W - 1 : px);
        int4 iv; iv.x = b; iv.y = py; iv.z = px; iv.w = qi;
        *(int4*)(indices + ((size_t)n * (PATCH * PATCH) + p) * 4) = iv;
    }

    // logits: serial over 49 positions, wave-parallel over D
    for (int p = 0; p < PATCH * PATCH; ++p) {
        int dy = p / PATCH - PATCH / 2;
        int dx = p % PATCH - PATCH / 2;
        int py = cy + dy; py = py < 0 ? 0 : (py > H - 1 ? H - 1 : py);
        int px = cx + dx; px = px < 0 ? 0 : (px > W - 1 ? W - 1 : px);
        const float* frow = fm + fm_b + ((size_t)py * W + px) * (size_t)D;

        float s = 0.0f;
        int c = 0;
#pragma unroll
        for (int d = lane * 4; d < D; d += 128, ++c) {
            float4 f = *(const float4*)(frow + d);
            float4 q = qreg[c];
            s += f.x * q.x + f.y * q.y + f.z * q.z + f.w * q.w;
        }
#pragma unroll
        for (int off = 16; off > 0; off >>= 1)
            s += __shfl_xor(s, off, 32);
        if (lane == 0) logits[(size_t)n * (PATCH * PATCH) + p] = s;
    }
}

extern "C" void kernel_launch(void* const* d_in, const int* in_sizes, int n_in,
                              void* d_out, int out_size, void* d_ws, size_t ws_size,
                              hipStream_t stream)
{
    (void)n_in; (void)out_size; (void)ws_size;

    const float* fm      = (const float*)d_in[0];
    const float* queries = (const float*)d_in[1];
    const float* qpos    = (const float*)d_in[2];
    // d_in[3] = query_batch_offsets (uniform Q; derived below instead)
    const int*   shapes  = (const int*)d_in[4];
    const float* Wts[4]  = {(const float*)d_in[5], (const float*)d_in[7],
                            (const float*)d_in[9], (const float*)d_in[11]};
    const float* bs[4]   = {(const float*)d_in[6], (const float*)d_in[8],
                            (const float*)d_in[10], (const float*)d_in[12]};

    const int D    = in_sizes[6];          // |b0|
    const int Ntot = in_sizes[1] / D;      // |queries| / D
    const int B    = in_sizes[4] / 2;      // |image_spatial_shapes| / 2
    const int Q    = Ntot / B;

    float* ws0 = (float*)d_ws;
    float* ws1 = ws0 + (size_t)Ntot * D;

    // ---- MLP: 4 WMMA GEMM layers, ping-pong through workspace ----
    const int tilesM  = (Ntot + 15) / 16;
    const int tilesN  = D / 64;            // 16x64 strip per wave
    const int waves   = tilesM * tilesN;
    const int threads = 256;
    const int gblocks = (waves * 32 + threads - 1) / threads;

    const float* in = queries;
    float* outs[4] = {ws0, ws1, ws0, ws1};
    for (int l = 0; l < 4; ++l) {
        const int relu = (l < 3) ? 1 : 0;
        if (D == 256) {
            mlp_layer_wmma<256, 256><<<gblocks, threads, 0, stream>>>(
                in, Wts[l], bs[l], outs[l], Ntot, D, D, relu);
        } else {
            mlp_layer_wmma<0, 0><<<gblocks, threads, 0, stream>>>(
                in, Wts[l], bs[l], outs[l], Ntot, D, D, relu);
        }
        in = outs[l];
    }

    // ---- Patch scoring + COO indices ----
    float* logits  = (float*)d_out;
    int*   indices = (int*)(logits + (size_t)Ntot * PATCH * PATCH);
    const int pblocks = (Ntot * 32 + threads - 1) / threads;
    if (D == 256) {
        patch_kernel<256><<<pblocks, threads, 0, stream>>>(
            fm, outs[3], qpos, shapes, logits, indices, B, Q, D);
    } else {
        patch_kernel<0><<<pblocks, threads, 0, stream>>>(
            fm, outs[3], qpos, shapes, logits, indices, B, Q, D);
    }
}